// DeepLPSI_63763084476519
// MI455X (gfx1250) — compile-verified
//
#include <hip/hip_runtime.h>

typedef float v2f __attribute__((ext_vector_type(2)));
typedef float v8f __attribute__((ext_vector_type(8)));

#define LN_EPS 1e-5f

// ---- wave32 8-lane xor reduction via DPP (no LDS) ----
// quad_perm(1,0,3,2)=0xB1 (xor1), quad_perm(2,3,0,1)=0x4E (xor2),
// ROW_HALF_MIRROR=0x141 (i^7 within 8-lane group; valid once quad-uniform)
__device__ __forceinline__ float dpp_sum8(float x) {
    float a = x + __int_as_float(__builtin_amdgcn_update_dpp(
                      0, __float_as_int(x), 0xB1, 0xF, 0xF, true));
    float b = a + __int_as_float(__builtin_amdgcn_update_dpp(
                      0, __float_as_int(a), 0x4E, 0xF, 0xF, true));
    float c = b + __int_as_float(__builtin_amdgcn_update_dpp(
                      0, __float_as_int(b), 0x141, 0xF, 0xF, true));
    return c;
}

// ---------------- node kernels ----------------

__global__ void zero_deg_kernel(float* __restrict__ a, float* __restrict__ b, int n) {
    int i = blockIdx.x * blockDim.x + threadIdx.x;
    if (i < n) { a[i] = 0.f; b[i] = 0.f; }
}

// degrees via float atomics (counts < 2^24 are exact in f32)
__global__ void degree_kernel(const int* __restrict__ src, const int* __restrict__ dst,
                              float* __restrict__ outd, float* __restrict__ ind, int E) {
    int t = blockIdx.x * blockDim.x + threadIdx.x;
    int e4 = E >> 2;
    if (t < e4) {
        int4 s = ((const int4*)src)[t];
        int4 d = ((const int4*)dst)[t];
        atomicAdd(&outd[s.x], 1.f); atomicAdd(&outd[s.y], 1.f);
        atomicAdd(&outd[s.z], 1.f); atomicAdd(&outd[s.w], 1.f);
        atomicAdd(&ind[d.x], 1.f);  atomicAdd(&ind[d.y], 1.f);
        atomicAdd(&ind[d.z], 1.f);  atomicAdd(&ind[d.w], 1.f);
    }
    if (t == 0) {
        for (int k = e4 << 2; k < E; ++k) {
            atomicAdd(&outd[src[k]], 1.f);
            atomicAdd(&ind[dst[k]], 1.f);
        }
    }
}

// finalize degree norms, sign-fix y, produce first messages, zero accumulator
__global__ void init_kernel(const float* __restrict__ y_in, float* __restrict__ ycur,
                            float* __restrict__ dsrc, float* __restrict__ ddst,
                            float* __restrict__ msg, float* __restrict__ acc, int n) {
    int i = blockIdx.x * blockDim.x + threadIdx.x;
    if (i < n) {
        float ds = rsqrtf(fmaxf(dsrc[i], 1.f));   // dsrc currently holds out-degree
        float dd = rsqrtf(fmaxf(ddst[i], 1.f));   // ddst currently holds in-degree
        dsrc[i] = ds;
        ddst[i] = dd;
        float yv = y_in[i];
        yv = (yv == 0.f) ? -1.f : yv;
        ycur[i] = yv;
        msg[i]  = yv * ds;
        acc[i]  = 0.f;
    }
}

// ---------------- edge kernel: acc[dst] += msg[src] ----------------

__global__ void edge_agg_kernel(const int* __restrict__ src, const int* __restrict__ dst,
                                const float* __restrict__ msg, float* __restrict__ acc, int E) {
    int t = blockIdx.x * blockDim.x + threadIdx.x;
    int e4 = E >> 2;
    if (t < e4) {
        int4 s = ((const int4*)src)[t];
        int4 d = ((const int4*)dst)[t];
        float m0 = msg[s.x], m1 = msg[s.y], m2 = msg[s.z], m3 = msg[s.w];
        atomicAdd(&acc[d.x], m0); atomicAdd(&acc[d.y], m1);
        atomicAdd(&acc[d.z], m2); atomicAdd(&acc[d.w], m3);
    }
    if (t == 0) {
        for (int k = e4 << 2; k < E; ++k) atomicAdd(&acc[dst[k]], msg[src[k]]);
    }
}

// ---------------- WMMA mid kernel ----------------
// One wave32 handles 16 nodes.
//   A (16x4 f32): row m = [agg[m]*ddst[m], y[m], ddst[m], 0]
//   B (4x16 f32): rows  = [W1; Wres; b1; 0], columns 8..15 = 0
//   D (16x16)   : D[m,j] = (agg*W1[j] + b1[j])*ddst + y*Wres[j]  for j<8
// Layernorm(8) + prelu via DPP 8-lane reductions; each lane captures its own
// node's final dot(vn, W2) so the msg/acc writes are one coalesced store pair.
__global__ void __launch_bounds__(256)
mid_kernel(float* __restrict__ acc, const float* __restrict__ ycur,
           const float* __restrict__ dsrc, const float* __restrict__ ddst,
           const float* __restrict__ W1, const float* __restrict__ b1,
           const float* __restrict__ Wres, const float* __restrict__ ln_g,
           const float* __restrict__ ln_b, const float* __restrict__ prelu_a,
           const float* __restrict__ W2, const float* __restrict__ b2,
           float* __restrict__ msg, int n) {
    int lane = (int)(threadIdx.x & 31);
    int wave = (int)((blockIdx.x * blockDim.x + threadIdx.x) >> 5);
    int base = wave * 16;
    if (base >= n) return;                 // wave-uniform exit (EXEC stays all-ones)

    int j    = lane & 15;                  // column index within half-wave
    int half = lane >> 4;                  // 0 or 1
    int jc   = j & 7;                      // clamped hidden index for param loads
    bool jlo = (j < 8);

    int nl = base + j; if (nl > n - 1) nl = n - 1;   // clamped node for A loads
    float aggv = acc[nl];
    float ddv  = ddst[nl];
    float yv   = ycur[nl];

    // A VGPR layout (32-bit A 16x4): v0 = K0 (lanes 0-15) / K2 (lanes 16-31),
    //                                v1 = K1 / K3
    float a0 = half ? ddv : aggv * ddv;
    float a1 = half ? 0.f : yv;

    // B VGPR layout (4x16): v0 = row0 / row2, v1 = row1 / row3
    float w1j = W1[jc], wrj = Wres[jc], b1j = b1[jc];
    w1j = jlo ? w1j : 0.f;  wrj = jlo ? wrj : 0.f;  b1j = jlo ? b1j : 0.f;
    float b0  = half ? b1j : w1j;
    float b1r = half ? 0.f : wrj;

    v2f A = {a0, a1};
    v2f B = {b0, b1r};
    v8f C = {0.f, 0.f, 0.f, 0.f, 0.f, 0.f, 0.f, 0.f};
    v8f D = __builtin_amdgcn_wmma_f32_16x16x4_f32(false, A, false, B, (short)0, C,
                                                  false, false);

    float gj  = ln_g[jc];
    float bj  = ln_b[jc];
    float pa  = prelu_a[0];
    float b2v = b2[0];
    float w2j = W2[jc];                    // unconditional load, then mask
    w2j = jlo ? w2j : 0.f;

    float tsel = 0.f;
    int   rsel = lane & 7;
    #pragma unroll
    for (int r = 0; r < 8; ++r) {
        float v  = D[r];                   // node (base + 8*half + r), hidden j
        float s1 = dpp_sum8(v);
        float s2 = dpp_sum8(v * v);
        float mu  = s1 * 0.125f;
        float var = s2 * 0.125f - mu * mu;
        float inv = rsqrtf(var + LN_EPS);
        float vn  = (v - mu) * inv * gj + bj;
        vn = (vn >= 0.f) ? vn : pa * vn;   // prelu
        float t = dpp_sum8(vn * w2j);      // full sum in every lane of the 8-group
        tsel = (r == rsel) ? t : tsel;     // lane (16*half + r) keeps its node's t
    }

    // lanes 0-7 own nodes base+0..7, lanes 16-23 own base+8..15
    int node = base + half * 8 + (lane & 7);
    if (((lane & 8) == 0) && node < n) {
        msg[node] = dsrc[node] * tsel + b2v;   // x2 message for second edge pass
        acc[node] = 0.f;                        // reset accumulator
    }
}

// ---------------- post kernel: y += acc*ddst; prep next iteration ----------------

__global__ void post_kernel(float* __restrict__ ycur, float* __restrict__ acc,
                            const float* __restrict__ ddst, const float* __restrict__ dsrc,
                            float* __restrict__ msg, int n, int last) {
    int i = blockIdx.x * blockDim.x + threadIdx.x;
    if (i < n) {
        float yv = ycur[i] + acc[i] * ddst[i];
        ycur[i] = yv;
        if (!last) {
            msg[i] = yv * dsrc[i];
            acc[i] = 0.f;
        }
    }
}

// ---------------- launch ----------------

extern "C" void kernel_launch(void* const* d_in, const int* in_sizes, int n_in,
                              void* d_out, int out_size, void* d_ws, size_t ws_size,
                              hipStream_t stream) {
    const float* y_in    = (const float*)d_in[0];
    const int*   src     = (const int*)  d_in[1];
    const int*   dst     = (const int*)  d_in[2];
    const float* W1      = (const float*)d_in[3];
    const float* b1      = (const float*)d_in[4];
    const float* Wres    = (const float*)d_in[5];
    const float* ln_g    = (const float*)d_in[6];
    const float* ln_b    = (const float*)d_in[7];
    const float* prelu_a = (const float*)d_in[8];
    const float* W2      = (const float*)d_in[9];
    const float* b2      = (const float*)d_in[10];

    int N = in_sizes[0];
    int E = in_sizes[1];

    float* out  = (float*)d_out;           // running y lives in d_out
    float* dsrc = (float*)d_ws;            // N floats
    float* ddst = dsrc + N;                // N floats
    float* msg  = ddst + N;                // N floats
    float* acc  = msg + N;                 // N floats

    int nb_n = (N + 255) / 256;
    int e4   = E >> 2;
    int nb_e = (e4 + 255) / 256; if (nb_e == 0) nb_e = 1;
    long long waves = ((long long)N + 15) / 16;
    int nb_m = (int)((waves * 32 + 255) / 256);

    zero_deg_kernel<<<nb_n, 256, 0, stream>>>(dsrc, ddst, N);
    degree_kernel  <<<nb_e, 256, 0, stream>>>(src, dst, dsrc, ddst, E);
    init_kernel    <<<nb_n, 256, 0, stream>>>(y_in, out, dsrc, ddst, msg, acc, N);

    for (int it = 0; it < 4; ++it) {
        edge_agg_kernel<<<nb_e, 256, 0, stream>>>(src, dst, msg, acc, E);
        mid_kernel     <<<nb_m, 256, 0, stream>>>(acc, out, dsrc, ddst,
                                                  W1, b1, Wres, ln_g, ln_b,
                                                  prelu_a, W2, b2, msg, N);
        edge_agg_kernel<<<nb_e, 256, 0, stream>>>(src, dst, msg, acc, E);
        post_kernel    <<<nb_n, 256, 0, stream>>>(out, acc, ddst, dsrc, msg, N, it == 3);
    }
}